// BlockDCTQuantizer_29051158790345
// MI455X (gfx1250) — compile-verified
//
#include <hip/hip_runtime.h>

typedef __attribute__((ext_vector_type(2))) float v2f;
typedef __attribute__((ext_vector_type(8))) float v8f;

#define IMG_W 1024
#define IMG_H 1024
#define LDS_STRIDE 20  // floats: conflict-free column reads (20r mod 64 distinct, r=0..15)

// cos(k*pi/16), k = 0..8
__device__ __constant__ float COS_PI_16[9] = {
    1.0f,
    0.98078528040323044f, 0.92387953251128674f, 0.83146961230254524f,
    0.70710678118654752f, 0.55557023301960222f, 0.38268343236508977f,
    0.19509032201612827f, 0.0f};

// Entry (i,j) of the 16x16 block-diagonal of the 8x8 orthonormal DCT-II matrix.
// D[i][j] = s(i) * cos(pi*i*(2j+1)/16), folded into the 9-entry quarter-wave table.
__device__ __forceinline__ float dblk_entry(int i, int j) {
    if ((i & 8) != (j & 8)) return 0.0f;           // off-diagonal 8x8 blocks are zero
    int ii = i & 7, jj = j & 7;
    float sg = (ii == 0) ? 0.35355339059327373f : 0.5f;  // sqrt(1/8), sqrt(2/8)
    int m = (ii * (2 * jj + 1)) & 31;              // cos(pi*m/16), periodic 32
    if (m > 16) m = 32 - m;                        // cos symmetric about pi
    if (m > 8) { sg = -sg; m = 16 - m; }           // cos(pi - t) = -cos(t)
    return sg * COS_PI_16[m];
}

// 16x16 = (16x16)*(16x16) as four accumulating K=4 fp32 WMMAs.
__device__ __forceinline__ v8f wmma_16x16(const v2f a[4], const v2f b[4]) {
    v8f c = {};
#pragma unroll
    for (int k = 0; k < 4; ++k)
        c = __builtin_amdgcn_wmma_f32_16x16x4_f32(false, a[k], false, b[k],
                                                  (short)0, c, false, false);
    return c;
}

// Load stage-1 B operands (rows of X) straight from global into WMMA B layout:
// lane (n,h) slice k holds X[4k+2h, n], X[4k+2h+1, n] -> 64B-coalesced row segments.
__device__ __forceinline__ void load_b_rows(const float* __restrict__ p, size_t base,
                                            int n, int h, v2f b[4]) {
#pragma unroll
    for (int k = 0; k < 4; ++k) {
        int r = 4 * k + 2 * h;
        b[k].x = p[base + (size_t)r * IMG_W + n];
        b[k].y = p[base + (size_t)(r + 1) * IMG_W + n];
    }
}

__device__ __forceinline__ size_t tile_base(int t) {
    int plane = t >> 12;                 // 64x64 tiles per 1024x1024 plane
    int rem = t & 4095;
    int ty = rem >> 6, tx = rem & 63;
    return (size_t)plane * (IMG_H * IMG_W) + (size_t)(ty * 16) * IMG_W + (size_t)(tx * 16);
}

__global__ __launch_bounds__(256) void
BlockDCTQuantizer_kernel(const float* __restrict__ x, float* __restrict__ out,
                         const int* __restrict__ qp_ptr, int tiles_total) {
    __shared__ float lds_all[8][16 * LDS_STRIDE];
    const int lane = threadIdx.x & 31;
    const int wave = threadIdx.x >> 5;
    float* sm = lds_all[wave];          // per-wave private tile; same-wave DS is in-order

    const int n = lane & 15;            // column (B/C layouts) or row (col reads)
    const int h = lane >> 4;            // lane-half select

    // A-operand lane constants (16x4 fp32 A layout: cols 4k+2h, 4k+2h+1 of row n).
    v2f aD[4], aDT[4];
#pragma unroll
    for (int k = 0; k < 4; ++k) {
        int c0 = 4 * k + 2 * h;
        aD[k].x  = dblk_entry(n, c0);      // M = diag(D, D)
        aD[k].y  = dblk_entry(n, c0 + 1);
        aDT[k].x = dblk_entry(c0, n);      // M^T
        aDT[k].y = dblk_entry(c0 + 1, n);
    }

    const float qstep = exp2f(((float)qp_ptr[0] - 4.0f) * (1.0f / 6.0f));
    const float inv_q = 1.0f / qstep;

    const int wave_stride = gridDim.x * 8;
    const int t0 = blockIdx.x * 8 + wave;

    v2f b[4];
    if (t0 < tiles_total) load_b_rows(x, tile_base(t0), n, h, b);

    for (int t = t0; t < tiles_total; t += wave_stride) {
        const size_t base = tile_base(t);

        // Prefetch next tile's stage-1 B operands (hidden behind 16 WMMAs + LDS work).
        v2f bn[4];
        const int tn = t + wave_stride;
        if (tn < tiles_total) load_b_rows(x, tile_base(tn), n, h, bn);

        v8f c;

        // Stage 1: T1 = M * X   (B came straight from global)
        c = wmma_16x16(aD, b);
#pragma unroll
        for (int r = 0; r < 8; ++r) sm[(r + 8 * h) * LDS_STRIDE + n] = c[r];

        // Stage 2: coeff^T = M * T1^T   (B-slices of T1^T = columns of T1)
#pragma unroll
        for (int k = 0; k < 4; ++k) {
            int cc = 4 * k + 2 * h;
            b[k].x = sm[n * LDS_STRIDE + cc];
            b[k].y = sm[n * LDS_STRIDE + cc + 1];
        }
        c = wmma_16x16(aD, b);

        // Quantize / dequantize; rintf == round-half-even == jnp.round
#pragma unroll
        for (int r = 0; r < 8; ++r) c[r] = rintf(c[r] * inv_q) * qstep;
#pragma unroll
        for (int r = 0; r < 8; ++r) sm[(r + 8 * h) * LDS_STRIDE + n] = c[r];

        // Stage 3: T3 = M^T * W   (row reads)
#pragma unroll
        for (int k = 0; k < 4; ++k) {
            int r = 4 * k + 2 * h;
            b[k].x = sm[r * LDS_STRIDE + n];
            b[k].y = sm[(r + 1) * LDS_STRIDE + n];
        }
        c = wmma_16x16(aDT, b);
#pragma unroll
        for (int r = 0; r < 8; ++r) sm[(r + 8 * h) * LDS_STRIDE + n] = c[r];

        // Stage 4: recon = M^T * T3^T   (column reads)
#pragma unroll
        for (int k = 0; k < 4; ++k) {
            int cc = 4 * k + 2 * h;
            b[k].x = sm[n * LDS_STRIDE + cc];
            b[k].y = sm[n * LDS_STRIDE + cc + 1];
        }
        c = wmma_16x16(aDT, b);

        // Store result directly from C layout: lane holds recon[r+8h, n].
#pragma unroll
        for (int r = 0; r < 8; ++r)
            out[base + (size_t)(r + 8 * h) * IMG_W + n] = c[r];

#pragma unroll
        for (int k = 0; k < 4; ++k) b[k] = bn[k];
    }
}

extern "C" void kernel_launch(void* const* d_in, const int* in_sizes, int n_in,
                              void* d_out, int out_size, void* d_ws, size_t ws_size,
                              hipStream_t stream) {
    const float* x = (const float*)d_in[0];
    // d_in[1] = block_size (specialized to 8), d_in[2] = qp
    const int* qp = (const int*)d_in[2];
    float* out = (float*)d_out;

    int planes = in_sizes[0] / (IMG_H * IMG_W);    // B*C = 24
    int tiles = planes * (IMG_H / 16) * (IMG_W / 16);

    // 8 waves per block, 4 tiles per wave via grid-stride loop (prefetch-pipelined)
    int blocks = (tiles + 8 * 4 - 1) / (8 * 4);
    if (blocks < 1) blocks = 1;
    BlockDCTQuantizer_kernel<<<blocks, 256, 0, stream>>>(x, out, qp, tiles);
}